// ForwardDCSim_4535485464791
// MI455X (gfx1250) — compile-verified
//
#include <hip/hip_runtime.h>

// ---------------------------------------------------------------------------
// DC resistivity forward: batched-16-RHS CG on a 7-point stencil, fp32.
// z-tridiagonal band applied with V_WMMA_F32_16X16X4_F32 (16 nodes x 16 src
// tiles, A-band precomputed in WMMA register layout); off-plane diagonals via
// VALU FMAs with packed float4 coefficients. Whole problem is L2-resident on
// MI455X (~52MB << 192MB L2).
// ---------------------------------------------------------------------------

typedef float v2f_t __attribute__((ext_vector_type(2)));
typedef float v8f_t __attribute__((ext_vector_type(8)));

// Problem geometry
#define NNY 65
#define NNZc 33
#define N_REAL 139425            // 65*65*33
#define NP     139440            // padded to 16-node tiles
#define TILES  8715
#define HALO   2160              // >= x-offset (2145) + 15, multiple of 16
#define OFFY   33                // flat y-neighbor offset
#define OFFX   2145              // flat x-neighbor offset (65*33)
#define NSRC   16
#define N_CELLS 131072           // 64*64*32
#define N_ACTIVE 81920           // 64*64*20
#define CG_ITERS 100
#define EPS_SHIFT 1.0e-7f
#define QW 6.25f                 // 0.25 * D (D = 25 m)

#define VLEN ((size_t)(NP + 2*HALO) * NSRC)   // floats per CG vector (w/ halo)

// ---------------------------------------------------------------- sigma cells
__global__ void k_sigma(const float* __restrict__ model, float* __restrict__ sigma) {
  int c = blockIdx.x * 256 + threadIdx.x;          // cell id = (i*64+j)*32+k
  int k = c & 31;
  int j = (c >> 5) & 63;
  int i = c >> 11;
  float s;
  if (k < 20) {                                    // active cell layers
    int aidx = (i * 64 + j) * 20 + k;
    s = 1.0f / model[aidx];
  } else {
    s = 1.0e-8f;                                   // air: 1/1e8
  }
  sigma[c] = s;
}

// --------------------------------------------------------- edge coefficients
__device__ __forceinline__ float sig_at(const float* sg, int ci, int cj, int ck) {
  if ((unsigned)ci < 64u && (unsigned)cj < 64u && (unsigned)ck < 32u)
    return sg[(ci * 64 + cj) * 32 + ck];
  return 0.0f;
}

__global__ void k_coeff(const float* __restrict__ sigma,
                        float4* __restrict__ coeff4,   // {cxm,cxp,cym,cyp} per node
                        float* __restrict__ czm, float* __restrict__ czp,
                        float* __restrict__ dia) {
  int n = blockIdx.x * 256 + threadIdx.x;
  if (n >= NP) return;
  if (n >= N_REAL) {                               // tail pad nodes
    coeff4[n] = make_float4(0.f, 0.f, 0.f, 0.f);
    czm[n] = 0.f; czp[n] = 0.f; dia[n] = EPS_SHIFT;
    return;
  }
  int k = n % NNZc;
  int rem = n / NNZc;
  int j = rem % NNY;
  int i = rem / NNY;
  // 8 surrounding cells s[a][b][c] = sigma(i-1+a, j-1+b, k-1+c), OOB -> 0
  float s000 = sig_at(sigma, i-1, j-1, k-1), s001 = sig_at(sigma, i-1, j-1, k);
  float s010 = sig_at(sigma, i-1, j,   k-1), s011 = sig_at(sigma, i-1, j,   k);
  float s100 = sig_at(sigma, i,   j-1, k-1), s101 = sig_at(sigma, i,   j-1, k);
  float s110 = sig_at(sigma, i,   j,   k-1), s111 = sig_at(sigma, i,   j,   k);
  float xm = QW * (s000 + s001 + s010 + s011);
  float xp = QW * (s100 + s101 + s110 + s111);
  float ym = QW * (s000 + s001 + s100 + s101);
  float yp = QW * (s010 + s011 + s110 + s111);
  float zm = QW * (s000 + s010 + s100 + s110);
  float zp = QW * (s001 + s011 + s101 + s111);
  coeff4[n] = make_float4(xm, xp, ym, yp);
  czm[n] = zm; czp[n] = zp;
  dia[n] = EPS_SHIFT + xm + xp + ym + yp + zm + zp;
}

// -------------------------- banded A tiles, stored in WMMA-A register layout
// aband[tile*320 + q*64 + lane*2 + {0,1}] = A(M, 4q+2h+{0,1}),  M=lane&15,
// h=lane>>4, band: A(M,M)=-czm, A(M,M+1)=diag, A(M,M+2)=-czp (window col = z-1+c)
__global__ void k_aband(const float* __restrict__ czm, const float* __restrict__ czp,
                        const float* __restrict__ dia, float* __restrict__ aband) {
  int t = blockIdx.x * 256 + threadIdx.x;          // t = tile*32 + lane
  if (t >= TILES * 32) return;
  const int lane = t & 31;
  const int m = t >> 5;
  const int h = lane >> 4;
  const int M = lane & 15;
  const float zL = czm[m * 16 + M];
  const float zU = czp[m * 16 + M];
  const float dg = dia[m * 16 + M];
  float* ab = aband + (size_t)m * 320 + lane * 2;
#pragma unroll
  for (int q = 0; q < 5; ++q) {
    const int d0 = 4 * q + 2 * h - M;
    const int d1 = d0 + 1;
    ab[q * 64]     = (d0 == 1) ? dg : (d0 == 0) ? -zL : (d0 == 2) ? -zU : 0.0f;
    ab[q * 64 + 1] = (d1 == 1) ? dg : (d1 == 0) ? -zL : (d1 == 2) ? -zU : 0.0f;
  }
}

// ----------------------------------------------------------------- CG init
__global__ void k_init(float* __restrict__ r, float* __restrict__ p,
                       const int* __restrict__ src_a, const int* __restrict__ src_b,
                       float* __restrict__ rs0) {
  int s = threadIdx.x;
  if (s < NSRC) {
    int a = src_a[s], b = src_b[s];
    r[a * NSRC + s] = 1.0f;
    r[b * NSRC + s] += -1.0f;
    p[a * NSRC + s] = 1.0f;
    p[b * NSRC + s] += -1.0f;
    rs0[s] = (a == b) ? 0.0f : 2.0f;               // ||r0||^2 per source
  }
}

// -------------------------------------------------- Ap = A*p  (WMMA + VALU)
__global__ void k_spmv(const float* __restrict__ p, float* __restrict__ Ap,
                       const float* __restrict__ aband,
                       const float4* __restrict__ coeff4,
                       float* __restrict__ pap_acc) {
  __shared__ float sdot[NSRC];
  if (threadIdx.x < NSRC) sdot[threadIdx.x] = 0.0f;
  __syncthreads();

  const int lane = threadIdx.x & 31;
  const int h = lane >> 4;                         // half-wave
  const int sN = lane & 15;                        // source column
  const int wid = blockIdx.x * (blockDim.x >> 5) + (threadIdx.x >> 5);
  const int nwaves = gridDim.x * (blockDim.x >> 5);

  float dotp = 0.0f;
  for (int m = wid; m < TILES; m += nwaves) {
    // ---- gather all WMMA fragments first (coalesced, immediate offsets)
    const float* ab = aband + (size_t)m * 320 + lane * 2;       // A tiles
    const float* pb = p + (m * 16 - 1 + 2 * h) * NSRC + sN;     // B window
    float ax[5], ay[5], bx[5], by[5];
#pragma unroll
    for (int q = 0; q < 5; ++q) {
      ax[q] = ab[q * 64];
      ay[q] = ab[q * 64 + 1];
      bx[q] = pb[q * 64];            // window row 4q+2h
      by[q] = pb[q * 64 + 16];       // window row 4q+2h+1
    }
    // ---- 5 back-to-back fp32 WMMAs: acc(16x16) = A_band(16x20) x W(20x16)
    v8f_t acc = {};
#pragma unroll
    for (int q = 0; q < 5; ++q) {
      v2f_t av; av.x = ax[q]; av.y = ay[q];
      v2f_t bv; bv.x = bx[q]; bv.y = by[q];
      acc = __builtin_amdgcn_wmma_f32_16x16x4_f32(
          false, av, false, bv, (short)0, acc, false, false);
    }

    // ---- off-plane diagonal terms + output + local dot (row M = v + 8h)
    const int nb = m * 16 + 8 * h;                 // first row of this half
    const float* pc  = p + nb * NSRC + sN;
    const float* pxm = pc - OFFX * NSRC;
    const float* pxp = pc + OFFX * NSRC;
    const float* pym = pc - OFFY * NSRC;
    const float* pyp = pc + OFFY * NSRC;
    const float4* c4 = coeff4 + nb;
    float* apb = Ap + nb * NSRC + sN;
#pragma unroll
    for (int v = 0; v < 8; ++v) {
      const float4 c = c4[v];
      float a = acc[v];
      a = fmaf(-c.x, pxm[v * NSRC], a);
      a = fmaf(-c.y, pxp[v * NSRC], a);
      a = fmaf(-c.z, pym[v * NSRC], a);
      a = fmaf(-c.w, pyp[v * NSRC], a);
      apb[v * NSRC] = a;
      dotp = fmaf(a, pc[v * NSRC], dotp);
    }
  }

  // per-source reduction: fold halves, LDS, then one global atomic per slot
  dotp += __shfl_xor(dotp, 16, 32);
  if (lane < NSRC) atomicAdd(&sdot[lane], dotp);
  __syncthreads();
  if (threadIdx.x < NSRC) atomicAdd(&pap_acc[threadIdx.x], sdot[threadIdx.x]);
}

// ------------------------------------------- x += a p ; r -= a Ap ; rs_new
__global__ void k_update(float4* __restrict__ x, float4* __restrict__ r,
                         const float4* __restrict__ p, const float4* __restrict__ Ap,
                         const float* __restrict__ rs_old,
                         const float* __restrict__ pap,
                         float* __restrict__ rs_new) {
  __shared__ float sdot[NSRC];
  if (threadIdx.x < NSRC) sdot[threadIdx.x] = 0.0f;
  __syncthreads();

  const int i = blockIdx.x * 256 + threadIdx.x;    // float4 index, [0, NP*4)
  if (i < NP * 4) {
    const int s0 = (i & 3) * 4;                    // sources s0..s0+3
    float al[4];
#pragma unroll
    for (int c = 0; c < 4; ++c) {
      const float pp = pap[s0 + c];
      al[c] = (pp != 0.0f) ? rs_old[s0 + c] / pp : 0.0f;
    }
    const float4 pv = p[i], ap = Ap[i];
    float4 xv = x[i], rv = r[i];
    xv.x = fmaf(al[0], pv.x, xv.x);  rv.x = fmaf(-al[0], ap.x, rv.x);
    xv.y = fmaf(al[1], pv.y, xv.y);  rv.y = fmaf(-al[1], ap.y, rv.y);
    xv.z = fmaf(al[2], pv.z, xv.z);  rv.z = fmaf(-al[2], ap.z, rv.z);
    xv.w = fmaf(al[3], pv.w, xv.w);  rv.w = fmaf(-al[3], ap.w, rv.w);
    x[i] = xv; r[i] = rv;
    atomicAdd(&sdot[s0 + 0], rv.x * rv.x);
    atomicAdd(&sdot[s0 + 1], rv.y * rv.y);
    atomicAdd(&sdot[s0 + 2], rv.z * rv.z);
    atomicAdd(&sdot[s0 + 3], rv.w * rv.w);
  }
  __syncthreads();
  if (threadIdx.x < NSRC) atomicAdd(&rs_new[threadIdx.x], sdot[threadIdx.x]);
}

// ------------------------------------------------------- p = r + beta * p
__global__ void k_pbeta(float4* __restrict__ p, const float4* __restrict__ r,
                        const float* __restrict__ rs_old,
                        const float* __restrict__ rs_new) {
  const int i = blockIdx.x * 256 + threadIdx.x;
  if (i >= NP * 4) return;
  const int s0 = (i & 3) * 4;
  float bt[4];
#pragma unroll
  for (int c = 0; c < 4; ++c) {
    const float ro = rs_old[s0 + c];
    bt[c] = (ro != 0.0f) ? rs_new[s0 + c] / ro : 0.0f;
  }
  const float4 rv = r[i];
  float4 pv = p[i];
  pv.x = fmaf(bt[0], pv.x, rv.x);
  pv.y = fmaf(bt[1], pv.y, rv.y);
  pv.z = fmaf(bt[2], pv.z, rv.z);
  pv.w = fmaf(bt[3], pv.w, rv.w);
  p[i] = pv;
}

// ------------------------------------------------------------ final losses
__global__ void k_final(const float* __restrict__ x,
                        const int* __restrict__ rxm, const int* __restrict__ rxn,
                        const float* __restrict__ dref,
                        const float* __restrict__ model,
                        const float* __restrict__ start,
                        float* __restrict__ out) {
  __shared__ float s_ld, s_lm;
  const int tid = threadIdx.x;
  if (tid == 0) { s_ld = 0.0f; s_lm = 0.0f; }
  __syncthreads();

  float pd = 0.0f;
  if (tid < 256) {                                 // dpred[s][rr], flat = tid
    const int s = tid >> 4;
    const int m = rxm[tid], n = rxn[tid];
    const float d = x[m * NSRC + s] - x[n * NSRC + s];
    const float e = d - dref[tid];
    pd = e * e;
  }
  float pm = 0.0f;
  for (int i = tid; i < N_ACTIVE; i += 1024) {
    const float dm = model[i] - start[i];
    pm = fmaf(dm, dm, pm);
  }
  atomicAdd(&s_ld, pd);
  atomicAdd(&s_lm, pm);
  __syncthreads();
  if (tid == 0) {
    const float ld = s_ld * (1.0f / 256.0f);
    const float lm = s_lm * (1.0f / (float)N_ACTIVE);
    out[0] = ld;                                   // alpha == 0
    out[1] = ld;
    out[2] = lm;
  }
}

// ---------------------------------------------------------------- launcher
extern "C" void kernel_launch(void* const* d_in, const int* in_sizes, int n_in,
                              void* d_out, int out_size, void* d_ws, size_t ws_size,
                              hipStream_t stream) {
  const float* model = (const float*)d_in[0];
  const float* start = (const float*)d_in[1];
  const float* dref  = (const float*)d_in[2];
  // d_in[3] = active_idx (layout derived analytically; unused)
  const int* src_a = (const int*)d_in[4];
  const int* src_b = (const int*)d_in[5];
  const int* rx_m  = (const int*)d_in[6];
  const int* rx_n  = (const int*)d_in[7];
  float* out = (float*)d_out;

  float* base = (float*)d_ws;
  const size_t OFF_X  = 0;
  const size_t OFF_R  = VLEN;
  const size_t OFF_P  = 2 * VLEN;
  const size_t OFF_AP = 3 * VLEN;
  const size_t OFF_C4 = 4 * VLEN;                  // NP*4 packed xy-coeffs
  const size_t OFF_CZ = OFF_C4 + (size_t)4 * NP;   // czm, czp, dia (3*NP)
  const size_t OFF_AB = OFF_CZ + (size_t)3 * NP;   // A-band tiles (20*NP)
  const size_t OFF_SG = OFF_AB + (size_t)20 * NP;  // sigma cells
  const size_t OFF_RS = OFF_SG + (size_t)N_CELLS;  // (CG_ITERS+1)*16
  const size_t OFF_PA = OFF_RS + (size_t)(CG_ITERS + 1) * NSRC; // CG_ITERS*16

  float*  xv  = base + OFF_X  + (size_t)HALO * NSRC;  // node-indexed bases
  float*  rv  = base + OFF_R  + (size_t)HALO * NSRC;
  float*  pv  = base + OFF_P  + (size_t)HALO * NSRC;
  float*  apv = base + OFF_AP + (size_t)HALO * NSRC;
  float4* c4  = (float4*)(base + OFF_C4);
  float*  czm = base + OFF_CZ + (size_t)0 * NP;
  float*  czp = base + OFF_CZ + (size_t)1 * NP;
  float*  dia = base + OFF_CZ + (size_t)2 * NP;
  float*  ab  = base + OFF_AB;
  float*  sigma = base + OFF_SG;
  float*  rs  = base + OFF_RS;
  float*  pap = base + OFF_PA;

  // zero the 4 CG vectors (incl. halos) and all per-iteration scalar slots
  hipMemsetAsync(base, 0, 4 * VLEN * sizeof(float), stream);
  hipMemsetAsync(base + OFF_RS, 0,
                 (size_t)(2 * CG_ITERS + 1) * NSRC * sizeof(float), stream);

  k_sigma<<<N_CELLS / 256, 256, 0, stream>>>(model, sigma);
  k_coeff<<<(NP + 255) / 256, 256, 0, stream>>>(sigma, c4, czm, czp, dia);
  k_aband<<<(TILES * 32 + 255) / 256, 256, 0, stream>>>(czm, czp, dia, ab);
  k_init<<<1, 32, 0, stream>>>(rv, pv, src_a, src_b, rs);

  const int EW_BLOCKS = (NP * 4 + 255) / 256;      // float4 element-wise grid
  for (int t = 0; t < CG_ITERS; ++t) {
    k_spmv<<<256, 256, 0, stream>>>(pv, apv, ab, c4, pap + (size_t)t * NSRC);
    k_update<<<EW_BLOCKS, 256, 0, stream>>>((float4*)xv, (float4*)rv,
                                            (const float4*)pv, (const float4*)apv,
                                            rs + (size_t)t * NSRC,
                                            pap + (size_t)t * NSRC,
                                            rs + (size_t)(t + 1) * NSRC);
    k_pbeta<<<EW_BLOCKS, 256, 0, stream>>>((float4*)pv, (const float4*)rv,
                                           rs + (size_t)t * NSRC,
                                           rs + (size_t)(t + 1) * NSRC);
  }

  k_final<<<1, 1024, 0, stream>>>(xv, rx_m, rx_n, dref, model, start, out);
}